// SparseDeConv3d_4415226380609
// MI455X (gfx1250) — compile-verified
//
#include <hip/hip_runtime.h>

typedef __attribute__((ext_vector_type(2))) float v2f;
typedef __attribute__((ext_vector_type(8))) float v8f;

#define N_IN   100000
#define N_OUT  200000
#define KOFF   27
#define INC    64
#define OUTC   64
#define EPS_BN 1e-5f

#define MTILES          ((N_IN + 15) / 16)   // 6250 16-row tiles
#define WAVES_PER_BLOCK 8
#define LDS_PITCH       66                   // padded pitch (even -> 8B aligned pairs)

// -------------------------------------------------------------------------
// Zero the output accumulator (must run first every call; harness poisons).
// -------------------------------------------------------------------------
__global__ void zero_f32(float* __restrict__ p, int n) {
    int i = blockIdx.x * blockDim.x + threadIdx.x;
    if (i < n) p[i] = 0.0f;
}

// -------------------------------------------------------------------------
// Gather -> WMMA f32 GEMM -> atomic scatter-add, one kernel offset per
// blockIdx.y, one 16-row m-tile per wave32.
// -------------------------------------------------------------------------
__global__ __launch_bounds__(256) void sparse_deconv_wmma_f32(
    const float* __restrict__ feats,    // [N_IN, INC]
    const float* __restrict__ weight,   // [KOFF, INC, OUTC]
    const int*   __restrict__ in_idx,   // [KOFF, N_IN]
    const int*   __restrict__ out_idx,  // [KOFF, N_IN]
    float*       __restrict__ out)      // [N_OUT, OUTC] accumulator
{
    __shared__ __align__(16) float wT[OUTC * LDS_PITCH]; // weight^T: wT[n][kk]

    const int k   = blockIdx.y;
    const int tid = threadIdx.x;

    // Cooperative stage of weight[k] (64x64 = 16KB) into LDS, transposed so a
    // (kk, kk+1) pair for a fixed output column n is one contiguous b64 read.
    const float* wk = weight + (size_t)k * (INC * OUTC);
    #pragma unroll
    for (int j = 0; j < (INC * OUTC) / 256; ++j) {
        int e  = tid + 256 * j;          // coalesced global read
        int kk = e >> 6;                 // row    (INC index)
        int n  = e & 63;                 // column (OUTC index)
        wT[n * LDS_PITCH + kk] = wk[e];
    }
    __syncthreads();

    const int wave = tid >> 5;
    const int lane = tid & 31;
    const int half = lane >> 4;          // 0 -> K pair (0,1), 1 -> K pair (2,3)
    const int l15  = lane & 15;

    const int mtile = blockIdx.x * WAVES_PER_BLOCK + wave;
    if (mtile >= MTILES) return;
    const int mbase = mtile * 16;

    // ---- A matrix (16x64): gathered rows of feats, ISA 32-bit A layout.
    // Lane L (L<16): M=L, holds K = 4c, 4c+1; lane L+16: same M, K = 4c+2, 4c+3.
    const int    arow = in_idx[(size_t)k * N_IN + mbase + l15];
    const float* ap   = feats + (size_t)arow * INC;

    v2f a[16];
    #pragma unroll
    for (int c = 0; c < 16; ++c)
        a[c] = *(const v2f*)(ap + 4 * c + 2 * half);   // global_load_b64

    // Output rows this lane scatters (D layout: VGPR r -> M = r + 8*half).
    int orow[8];
    #pragma unroll
    for (int r = 0; r < 8; ++r)
        orow[r] = out_idx[(size_t)k * N_IN + mbase + r + 8 * half];

    // ---- 4 N-tiles of 16 columns; 16 chained V_WMMA_F32_16X16X4_F32 each.
    #pragma unroll
    for (int nt = 0; nt < 4; ++nt) {
        const int    ncol = nt * 16 + l15;             // this lane's N column
        const float* bt   = &wT[ncol * LDS_PITCH];

        v8f acc = {};
        #pragma unroll
        for (int c = 0; c < 16; ++c) {
            v2f b = *(const v2f*)(bt + 4 * c + 2 * half);  // ds_load_b64
            acc = __builtin_amdgcn_wmma_f32_16x16x4_f32(
                /*neg_a=*/false, a[c], /*neg_b=*/false, b,
                /*c_mod=*/(short)0, acc, /*reuse_a=*/false, /*reuse_b=*/false);
        }

        #pragma unroll
        for (int r = 0; r < 8; ++r)
            atomicAdd(&out[(size_t)orow[r] * OUTC + ncol], acc[r]);
    }
}

// -------------------------------------------------------------------------
// Eval-mode BatchNorm + ReLU, in place over the accumulator.
// -------------------------------------------------------------------------
__global__ void bn_relu_inplace(float* __restrict__ out,
                                const float* __restrict__ gamma,
                                const float* __restrict__ beta,
                                const float* __restrict__ mean,
                                const float* __restrict__ var)
{
    int i = blockIdx.x * blockDim.x + threadIdx.x;
    if (i >= N_OUT * OUTC) return;
    int c = i & (OUTC - 1);
    float inv = rsqrtf(var[c] + EPS_BN);
    float y   = (out[i] - mean[c]) * (inv * gamma[c]) + beta[c];
    out[i]    = fmaxf(y, 0.0f);
}

// -------------------------------------------------------------------------
extern "C" void kernel_launch(void* const* d_in, const int* in_sizes, int n_in,
                              void* d_out, int out_size, void* d_ws, size_t ws_size,
                              hipStream_t stream) {
    const float* feats   = (const float*)d_in[0];
    const float* weight  = (const float*)d_in[1];
    const int*   in_idx  = (const int*)  d_in[2];
    const int*   out_idx = (const int*)  d_in[3];
    const float* gamma   = (const float*)d_in[4];
    const float* beta    = (const float*)d_in[5];
    const float* rmean   = (const float*)d_in[6];
    const float* rvar    = (const float*)d_in[7];
    float*       out     = (float*)d_out;

    const int total = N_OUT * OUTC;

    zero_f32<<<(total + 255) / 256, 256, 0, stream>>>(out, total);

    dim3 grid((MTILES + WAVES_PER_BLOCK - 1) / WAVES_PER_BLOCK, KOFF);
    sparse_deconv_wmma_f32<<<grid, 256, 0, stream>>>(feats, weight, in_idx,
                                                     out_idx, out);

    bn_relu_inplace<<<(total + 255) / 256, 256, 0, stream>>>(out, gamma, beta,
                                                             rmean, rvar);
}